// NEATWithMemory_64458869178769
// MI455X (gfx1250) — compile-verified
//
#include <hip/hip_runtime.h>
#include <hip/hip_bf16.h>

typedef __attribute__((ext_vector_type(4))) int   v4i;
typedef __attribute__((ext_vector_type(4))) float v4f;
typedef __attribute__((ext_vector_type(2))) float v2f;
typedef __attribute__((ext_vector_type(8))) float v8f;

#define OUT_N      512
#define THREADS    256
#define WAVES_PB   8          // 256 threads / wave32

// Branchless gather from the virtual concatenated input vector
// [terrain | population | memory | bias=1]. One global_load_b32 per lane,
// address picked with v_cndmask chains (no divergent branches).
__device__ __forceinline__ float gather_x(int si,
    const float* __restrict__ terr, const float* __restrict__ pop,
    const float* __restrict__ mem, int n1, int n2, int n3)
{
    const float* base = terr;
    int off = si;
    if (si >= n1) { base = pop; off = si - n1; }
    if (si >= n2) { base = mem; off = si - n2; }
    bool bias = (si >= n3);
    if (bias) off = 0;                 // safe address for the bias lane
    float x = base[off];
    return bias ? 1.0f : x;
}

// Hot loop: stream edges with NT b128 loads, scatter into per-wave LDS
// accumulators (ds_add_f32), then either write a per-block partial row to
// d_ws or atomically fold into d_out (fallback).
template <bool ATOMIC>
__global__ __launch_bounds__(THREADS)
void neat_edges_kernel(const float* __restrict__ terr,
                       const float* __restrict__ pop,
                       const float* __restrict__ mem,
                       const int*   __restrict__ src,
                       const int*   __restrict__ dst,
                       const float* __restrict__ w,
                       float*       __restrict__ outp,   // partials[NB][512] or d_out[512]
                       int n_edges, int n1, int n2, int n3)
{
    __shared__ float acc[WAVES_PB][OUT_N];

    const int tid  = threadIdx.x;
    const int wave = tid >> 5;

    #pragma unroll
    for (int i = tid; i < WAVES_PB * OUT_N; i += THREADS)
        ((float*)acc)[i] = 0.0f;
    __syncthreads();

    float* wacc = acc[wave];

    const v4i* __restrict__ src4 = (const v4i*)src;
    const v4i* __restrict__ dst4 = (const v4i*)dst;
    const v4f* __restrict__ w4   = (const v4f*)w;

    const int nquads = n_edges >> 2;
    for (int q = blockIdx.x * THREADS + tid; q < nquads;
         q += gridDim.x * THREADS) {
        v4i s  = __builtin_nontemporal_load(src4 + q);
        v4i d  = __builtin_nontemporal_load(dst4 + q);
        v4f ww = __builtin_nontemporal_load(w4  + q);
        #pragma unroll
        for (int k = 0; k < 4; ++k) {
            float x = gather_x(s[k], terr, pop, mem, n1, n2, n3);
            atomicAdd(&wacc[d[k] & (OUT_N - 1)], x * ww[k]);   // ds_add_f32
        }
    }

    // Tail (n_edges % 4) handled by block 0 only.
    if (blockIdx.x == 0) {
        for (int e = (nquads << 2) + tid; e < n_edges; e += THREADS) {
            float x = gather_x(src[e], terr, pop, mem, n1, n2, n3);
            atomicAdd(&wacc[dst[e] & (OUT_N - 1)], x * w[e]);
        }
    }
    __syncthreads();

    // Fold the 8 wave-private copies; emit this block's 512-float partial.
    for (int c = tid; c < OUT_N; c += THREADS) {
        float v = acc[0][c] + acc[1][c] + acc[2][c] + acc[3][c] +
                  acc[4][c] + acc[5][c] + acc[6][c] + acc[7][c];
        if (ATOMIC) atomicAdd(&outp[c], v);                    // global_atomic_add_f32
        else        outp[(size_t)blockIdx.x * OUT_N + c] = v;
    }
}

// Cross-block reduction via V_WMMA_F32_16X16X4_F32.
// out[n] = sum_b P[b][n].  A = ones(16x4)  =>  D[m][n] = sum_k B[k][n];
// each WMMA folds 4 partial rows for 16 columns; C accumulates over b.
// B layout (4x16, K=4): VGPR0 = K{0|2}, VGPR1 = K{1|3}; lanes 0-15 / 16-31
// carry the two K halves, column = lane & 15 (mirrors documented A/C layouts).
__global__ __launch_bounds__(THREADS)
void neat_reduce_wmma_kernel(const float* __restrict__ partials,
                             float* __restrict__ out, int nb)
{
    const int tid      = threadIdx.x;
    const int lane     = tid & 31;
    const int gwave    = (blockIdx.x * THREADS + tid) >> 5;
    const int colbase  = gwave * 16;
    if (colbase >= OUT_N) return;          // uniform per wave

    const int half = lane >> 4;            // which K-pair this lane holds
    const int col  = colbase + (lane & 15);

    v2f a; a.x = 1.0f; a.y = 1.0f;         // all-ones A matrix
    v8f c = {};
    for (int b = 0; b < nb; b += 4) {
        const float* row = partials + (size_t)(b + 2 * half) * OUT_N + col;
        v2f bm;
        bm.x = row[0];
        bm.y = row[OUT_N];
        c = __builtin_amdgcn_wmma_f32_16x16x4_f32(
                /*neg_a=*/false, a, /*neg_b=*/false, bm,
                /*c_mod=*/(short)0, c, /*reuse_a=*/false, /*reuse_b=*/false);
    }
    // D VGPR0: lanes 0-15 hold M=0, N=lane  (every row of D is the colsum)
    if (lane < 16) out[col] = c[0];
}

__global__ void zero_out_kernel(float* __restrict__ out, int n)
{
    int i = blockIdx.x * blockDim.x + threadIdx.x;
    if (i < n) out[i] = 0.0f;
}

extern "C" void kernel_launch(void* const* d_in, const int* in_sizes, int n_in,
                              void* d_out, int out_size, void* d_ws, size_t ws_size,
                              hipStream_t stream)
{
    const float* terr = (const float*)d_in[0];
    const float* pop  = (const float*)d_in[1];
    const float* mem  = (const float*)d_in[2];
    const int*   src  = (const int*)  d_in[3];
    const int*   dst  = (const int*)  d_in[4];
    const float* w    = (const float*)d_in[5];
    float* out = (float*)d_out;

    const int n1      = in_sizes[0];          // terrain length
    const int n2      = n1 + in_sizes[1];     // + population
    const int n3      = n2 + in_sizes[2];     // + memory  (== bias index)
    const int n_edges = in_sizes[3];

    // Size the partials grid from available workspace (multiple of 4 rows
    // for the WMMA k=4 reduction).
    int maxNB = (int)(ws_size / (OUT_N * sizeof(float)));
    maxNB &= ~3;

    if (maxNB >= 64) {
        const int NB = maxNB < 2048 ? maxNB : 2048;
        float* partials = (float*)d_ws;
        neat_edges_kernel<false><<<NB, THREADS, 0, stream>>>(
            terr, pop, mem, src, dst, w, partials, n_edges, n1, n2, n3);
        neat_reduce_wmma_kernel<<<(OUT_N / 16 + WAVES_PB - 1) / WAVES_PB,
                                  THREADS, 0, stream>>>(partials, out, NB);
    } else {
        // Workspace too small: zero output then accumulate with global atomics.
        zero_out_kernel<<<(out_size + THREADS - 1) / THREADS, THREADS, 0, stream>>>(
            out, out_size);
        neat_edges_kernel<true><<<1024, THREADS, 0, stream>>>(
            terr, pop, mem, src, dst, w, out, n_edges, n1, n2, n3);
    }
}